// GAT_11252814315553
// MI455X (gfx1250) — compile-verified
//
#include <hip/hip_runtime.h>
#include <math.h>

// ---------------------------------------------------------------------------
// GAT forward for MI455X (gfx1250, wave32).
//  - Dense GEMMs use V_WMMA_F32_16X16X4_F32 (fp32-exact matrix pipe).
//  - Graph aggregation is gather-based via on-device CSR (no float atomics,
//    deterministic, coalesced 1.74GB stream that is the roofline bottleneck).
// ---------------------------------------------------------------------------

typedef __attribute__((ext_vector_type(2))) float v2f;
typedef __attribute__((ext_vector_type(8))) float v8f;

#define NN 50000
#define EE 800000
#define GG 128

// ------------------------- WMMA fp32 GEMM ----------------------------------
// C[M,Nc] = A[M,K] @ B[K,Nc], row-major. blockDim=128 (4 waves), each wave
// owns one 16x16 tile; grid = (Nc/64, M/16). K % 4 == 0, M % 16 == 0.
__global__ void gat_gemm_wmma_f32(const float* __restrict__ A,
                                  const float* __restrict__ B,
                                  float* __restrict__ C,
                                  int K, int Nc) {
  const int wave = threadIdx.x >> 5;
  const int lane = threadIdx.x & 31;
  const int tileN = blockIdx.x * 4 + wave;
  const int tileM = blockIdx.y;
  const int row0 = tileM * 16;
  const int col0 = tileN * 16;
  const int half = lane >> 4;    // 0: lanes 0-15, 1: lanes 16-31
  const int l16  = lane & 15;

  v8f acc = {0.f, 0.f, 0.f, 0.f, 0.f, 0.f, 0.f, 0.f};

  const float* Arow = A + (size_t)(row0 + l16) * K;   // A: row = M index (l16)
  const float* Bcol = B + col0 + l16;                 // B: col = N index (l16)

#pragma unroll 4
  for (int k = 0; k < K; k += 4) {
    // A 16x4 layout: VGPR0 = K = k+2*half, VGPR1 = K = k+1+2*half
    v2f a, b;
    const int ka = k + 2 * half;
    a.x = Arow[ka];
    a.y = Arow[ka + 1];
    // B 4x16 layout mirrors A with M->N: rows K=k+2*half, K=k+1+2*half
    b.x = Bcol[(size_t)ka * Nc];
    b.y = Bcol[(size_t)(ka + 1) * Nc];
    acc = __builtin_amdgcn_wmma_f32_16x16x4_f32(false, a, false, b,
                                                (short)0, acc, false, false);
  }

  // C/D layout: VGPR j -> row (j + 8*half), col = l16
  float* Cp = C + (size_t)(row0 + half * 8) * Nc + col0 + l16;
#pragma unroll
  for (int j = 0; j < 8; ++j) Cp[(size_t)j * Nc] = acc[j];
}

// ------------------------- attention coefficients --------------------------
// a_src[n,h] = sum_c h[n, h*128+c] * att_s[h,c]  (and same for a_dst).
__global__ void gat_attn_coef(const float* __restrict__ h,
                              const float* __restrict__ att_s,
                              const float* __restrict__ att_d,
                              float* __restrict__ as, float* __restrict__ ad,
                              int heads, int total) {
  int idx = blockIdx.x * blockDim.x + threadIdx.x;   // idx = n*heads + hd
  if (idx >= total) return;
  int hd = idx % heads;
  const float* hp = h + (size_t)idx * 128;           // n*heads*128 + hd*128
  const float* ws = att_s + hd * 128;
  const float* wd = att_d + hd * 128;
  float s = 0.f, d = 0.f;
#pragma unroll 8
  for (int c = 0; c < 128; ++c) {
    float v = hp[c];
    s += v * ws[c];
    d += v * wd[c];
  }
  as[idx] = s;
  ad[idx] = d;
}

// ------------------------- CSR construction --------------------------------
__global__ void gat_init(int* __restrict__ deg, int* __restrict__ cursor) {
  int i = blockIdx.x * blockDim.x + threadIdx.x;
  if (i < NN) { deg[i] = 1; cursor[i] = 0; }   // deg starts at 1 (self loop)
}

__global__ void gat_count(const int* __restrict__ dst, int* __restrict__ deg) {
  int e = blockIdx.x * blockDim.x + threadIdx.x;
  if (e < EE) atomicAdd(&deg[dst[e]], 1);
}

// Single-block exclusive scan: start[0..n] from deg[0..n-1].
__global__ void gat_scan(const int* __restrict__ deg, int* __restrict__ start, int n) {
  __shared__ int sh[256];
  __shared__ int carry;
  const int tid = threadIdx.x;
  if (tid == 0) carry = 0;
  __syncthreads();
  for (int base = 0; base < n; base += 256) {
    int i = base + tid;
    int v = (i < n) ? deg[i] : 0;
    sh[tid] = v;
    __syncthreads();
    for (int off = 1; off < 256; off <<= 1) {
      int t = (tid >= off) ? sh[tid - off] : 0;
      __syncthreads();
      sh[tid] += t;
      __syncthreads();
    }
    if (i < n) start[i] = carry + (sh[tid] - v);
    __syncthreads();
    if (tid == 255) carry += sh[255];
    __syncthreads();
  }
  if (tid == 0) start[n] = carry;
}

__global__ void gat_fill(const int* __restrict__ dst, const int* __restrict__ start,
                         int* __restrict__ cursor, int* __restrict__ elist) {
  int e = blockIdx.x * blockDim.x + threadIdx.x;
  if (e < EE) {
    int d = dst[e];
    int pos = atomicAdd(&cursor[d], 1);
    elist[start[d] + pos] = e;
  }
}

__global__ void gat_selfloop(const int* __restrict__ start, const int* __restrict__ deg,
                             int* __restrict__ elist) {
  int n = blockIdx.x * blockDim.x + threadIdx.x;
  if (n < NN) elist[start[n] + deg[n] - 1] = EE + n;   // encode self loop
}

// ------------------------- layer-1 aggregation (H=4, C=128) ----------------
// blockDim = 256; thread t owns channels {t, t+256}.
__global__ void gat_agg1(const float* __restrict__ h, const float* __restrict__ asrc,
                         const float* __restrict__ adst, const int* __restrict__ srcIdx,
                         const int* __restrict__ start, const int* __restrict__ deg,
                         const int* __restrict__ elist, const float* __restrict__ bias,
                         float* __restrict__ out) {
  const int n = blockIdx.x;
  const int t = threadIdx.x;
  const int s0 = start[n];
  const int cnt = deg[n];
  __shared__ float s_stat[8];        // [0..3]=max per head, [4..7]=denom
  __shared__ float s_alpha[64 * 4];
  __shared__ int   s_src[64];

  const int head = t >> 5, lane = t & 31;
  if (t < 128) {
    const float ad = adst[n * 4 + head];
    float m = -3.0e38f;
    for (int i = lane; i < cnt; i += 32) {
      int e = elist[s0 + i];
      int s = (e < EE) ? srcIdx[e] : (e - EE);
      float v = asrc[s * 4 + head] + ad;
      v = v > 0.f ? v : 0.2f * v;                    // leaky_relu 0.2
      m = fmaxf(m, v);
    }
    for (int off = 16; off; off >>= 1) m = fmaxf(m, __shfl_xor(m, off));
    if (lane == 0) s_stat[head] = m;
  }
  __syncthreads();
  if (t < 128) {
    const float ad = adst[n * 4 + head];
    const float m = s_stat[head];
    float den = 0.f;
    for (int i = lane; i < cnt; i += 32) {
      int e = elist[s0 + i];
      int s = (e < EE) ? srcIdx[e] : (e - EE);
      float v = asrc[s * 4 + head] + ad;
      v = v > 0.f ? v : 0.2f * v;
      den += expf(v - m);
    }
    for (int off = 16; off; off >>= 1) den += __shfl_xor(den, off);
    if (lane == 0) s_stat[4 + head] = den;
  }
  __syncthreads();

  const int c0 = t, c1 = t + 256;
  const int h0 = c0 >> 7, h1 = c1 >> 7;
  const int ai = t & 63, ah = t >> 6;
  const float adh = adst[n * 4 + ah];
  const float mh = s_stat[ah];
  const float dh = s_stat[4 + ah] + 1e-16f;
  float acc0 = 0.f, acc1 = 0.f;

  for (int base = 0; base < cnt; base += 64) {
    int tile = cnt - base; if (tile > 64) tile = 64;
    if (ai < tile) {
      int e = elist[s0 + base + ai];
      int s = (e < EE) ? srcIdx[e] : (e - EE);
      if (ah == 0) s_src[ai] = s;
      float v = asrc[s * 4 + ah] + adh;
      v = v > 0.f ? v : 0.2f * v;
      s_alpha[ai * 4 + ah] = expf(v - mh) / dh;
    }
    __syncthreads();
    for (int i = 0; i < tile; ++i) {
      const float* hp = h + (size_t)s_src[i] * 512;
      acc0 += hp[c0] * s_alpha[i * 4 + h0];
      acc1 += hp[c1] * s_alpha[i * 4 + h1];
    }
    __syncthreads();
  }
  out[(size_t)n * 512 + c0] = acc0 + bias[c0];
  out[(size_t)n * 512 + c1] = acc1 + bias[c1];
}

// ------------------------- layer-2 aggregation (H=1, C=128) ----------------
__global__ void gat_agg2(const float* __restrict__ h, const float* __restrict__ asrc,
                         const float* __restrict__ adst, const int* __restrict__ srcIdx,
                         const int* __restrict__ start, const int* __restrict__ deg,
                         const int* __restrict__ elist, const float* __restrict__ bias,
                         float* __restrict__ out) {
  const int n = blockIdx.x;
  const int t = threadIdx.x;                  // 128
  const int s0 = start[n];
  const int cnt = deg[n];
  __shared__ float s_stat[2];                 // max, denom
  __shared__ float s_alpha[128];
  __shared__ int   s_src[128];
  const float ad = adst[n];

  if (t < 32) {
    float m = -3.0e38f;
    for (int i = t; i < cnt; i += 32) {
      int e = elist[s0 + i];
      int s = (e < EE) ? srcIdx[e] : (e - EE);
      float v = asrc[s] + ad;
      v = v > 0.f ? v : 0.2f * v;
      m = fmaxf(m, v);
    }
    for (int off = 16; off; off >>= 1) m = fmaxf(m, __shfl_xor(m, off));
    if (t == 0) s_stat[0] = m;
  }
  __syncthreads();
  if (t < 32) {
    const float m = s_stat[0];
    float den = 0.f;
    for (int i = t; i < cnt; i += 32) {
      int e = elist[s0 + i];
      int s = (e < EE) ? srcIdx[e] : (e - EE);
      float v = asrc[s] + ad;
      v = v > 0.f ? v : 0.2f * v;
      den += expf(v - m);
    }
    for (int off = 16; off; off >>= 1) den += __shfl_xor(den, off);
    if (t == 0) s_stat[1] = den;
  }
  __syncthreads();
  const float mh = s_stat[0];
  const float dh = s_stat[1] + 1e-16f;
  float acc = 0.f;

  for (int base = 0; base < cnt; base += 128) {
    int tile = cnt - base; if (tile > 128) tile = 128;
    if (t < tile) {
      int e = elist[s0 + base + t];
      int s = (e < EE) ? srcIdx[e] : (e - EE);
      s_src[t] = s;
      float v = asrc[s] + ad;
      v = v > 0.f ? v : 0.2f * v;
      s_alpha[t] = expf(v - mh) / dh;
    }
    __syncthreads();
    for (int i = 0; i < tile; ++i)
      acc += h[(size_t)s_src[i] * 128 + t] * s_alpha[i];
    __syncthreads();
  }
  out[(size_t)n * 128 + t] = acc + bias[t];
}

// ------------------------- layernorm + relu (in place) ---------------------
// blockDim = D/per with per in {1,2}; launched with blockDim 256 (D=512) or
// 128 (D=128).
__global__ void gat_ln_relu(float* __restrict__ x, const float* __restrict__ gamma,
                            const float* __restrict__ beta, int D) {
  const int n = blockIdx.x;
  const int t = threadIdx.x;
  const int B = blockDim.x;
  const int per = D / B;                      // 1 or 2
  const size_t base = (size_t)n * D;
  float v0 = x[base + t];
  float v1 = (per == 2) ? x[base + t + B] : 0.f;
  float s = v0 + v1, q = v0 * v0 + v1 * v1;
  __shared__ float rs[8], rq[8];
  for (int off = 16; off; off >>= 1) { s += __shfl_xor(s, off); q += __shfl_xor(q, off); }
  const int wid = t >> 5, lane = t & 31;
  if (lane == 0) { rs[wid] = s; rq[wid] = q; }
  __syncthreads();
  if (t == 0) {
    float ts = 0.f, tq = 0.f;
    const int nw = B >> 5;
    for (int w = 0; w < nw; ++w) { ts += rs[w]; tq += rq[w]; }
    rs[0] = ts; rq[0] = tq;
  }
  __syncthreads();
  const float mean = rs[0] / (float)D;
  const float var  = rq[0] / (float)D - mean * mean;
  const float rstd = rsqrtf(var + 1e-5f);
  float y0 = (v0 - mean) * rstd * gamma[t] + beta[t];
  x[base + t] = fmaxf(y0, 0.f);
  if (per == 2) {
    float y1 = (v1 - mean) * rstd * gamma[t + B] + beta[t + B];
    x[base + t + B] = fmaxf(y1, 0.f);
  }
}

// ------------------------- fused mean-pool + MLP head ----------------------
// batch is sorted -> one block (128 threads) per graph; binary search range.
__global__ void gat_pool_head(const float* __restrict__ h2, const int* __restrict__ batch,
                              const float* __restrict__ rW1, const float* __restrict__ rb1,
                              const float* __restrict__ rg, const float* __restrict__ rbe,
                              const float* __restrict__ rW2, const float* __restrict__ rb2,
                              float* __restrict__ out) {
  const int g = blockIdx.x;
  const int t = threadIdx.x;                  // 128
  __shared__ int s_lo, s_hi;
  __shared__ float p[128], r[32];
  if (t == 0) {
    int lo = 0, hi = NN;
    while (lo < hi) { int mid = (lo + hi) >> 1; if (batch[mid] <  g) lo = mid + 1; else hi = mid; }
    s_lo = lo;
    int lo2 = lo, hi2 = NN;
    while (lo2 < hi2) { int mid = (lo2 + hi2) >> 1; if (batch[mid] <= g) lo2 = mid + 1; else hi2 = mid; }
    s_hi = lo2;
  }
  __syncthreads();
  const int lo = s_lo, hi = s_hi;
  float acc = 0.f;
  for (int n = lo; n < hi; ++n) acc += h2[(size_t)n * 128 + t];
  p[t] = acc / fmaxf((float)(hi - lo), 1.0f);
  __syncthreads();
  if (t < 32) {
    float z = rb1[t];
    for (int k = 0; k < 128; ++k) z += p[k] * rW1[k * 32 + t];
    float s = z, q = z * z;
    for (int off = 16; off; off >>= 1) { s += __shfl_xor(s, off); q += __shfl_xor(q, off); }
    const float mu = s / 32.f;
    const float var = q / 32.f - mu * mu;
    float y = (z - mu) * rsqrtf(var + 1e-5f) * rg[t] + rbe[t];
    r[t] = fmaxf(y, 0.f);
  }
  __syncthreads();
  if (t < 8) {
    float o = rb2[t];
    for (int j = 0; j < 32; ++j) o += r[j] * rW2[j * 8 + t];
    out[g * 8 + t] = o;
  }
}

// ---------------------------------------------------------------------------
extern "C" void kernel_launch(void* const* d_in, const int* in_sizes, int n_in,
                              void* d_out, int out_size, void* d_ws, size_t ws_size,
                              hipStream_t stream) {
  const float* x       = (const float*)d_in[0];
  const int*   eidx    = (const int*)d_in[1];
  const int*   srcIdx  = eidx;         // edge_index[0]
  const int*   dstIdx  = eidx + EE;    // edge_index[1]
  // d_in[2] = edge_weight (unused by reference)
  const int*   batch   = (const int*)d_in[3];
  const float* W1      = (const float*)d_in[4];
  const float* as1w    = (const float*)d_in[5];
  const float* ad1w    = (const float*)d_in[6];
  const float* b1      = (const float*)d_in[7];
  const float* g1      = (const float*)d_in[8];
  const float* be1     = (const float*)d_in[9];
  const float* W2      = (const float*)d_in[10];
  const float* as2w    = (const float*)d_in[11];
  const float* ad2w    = (const float*)d_in[12];
  const float* b2      = (const float*)d_in[13];
  const float* g2      = (const float*)d_in[14];
  const float* be2     = (const float*)d_in[15];
  const float* rW1     = (const float*)d_in[16];
  const float* rb1     = (const float*)d_in[17];
  const float* rg      = (const float*)d_in[18];
  const float* rbe     = (const float*)d_in[19];
  const float* rW2     = (const float*)d_in[20];
  const float* rb2     = (const float*)d_in[21];
  float* out = (float*)d_out;

  // ---- workspace carve (~211 MB) ----
  char* wsc = (char*)d_ws;
  auto carve = [&](size_t bytes) {
    void* p = (void*)wsc;
    wsc += (bytes + 255) & ~(size_t)255;
    return p;
  };
  float* bufA  = (float*)carve((size_t)NN * 512 * sizeof(float)); // h1pre, later h2pre+agg2
  float* bufB  = (float*)carve((size_t)NN * 512 * sizeof(float)); // agg1 -> h1 (LN in place)
  float* as1   = (float*)carve((size_t)NN * 4 * sizeof(float));
  float* ad1   = (float*)carve((size_t)NN * 4 * sizeof(float));
  float* as2   = (float*)carve((size_t)NN * sizeof(float));
  float* ad2   = (float*)carve((size_t)NN * sizeof(float));
  int*   deg   = (int*)carve((size_t)NN * sizeof(int));
  int*   start = (int*)carve((size_t)(NN + 1) * sizeof(int));
  int*   curs  = (int*)carve((size_t)NN * sizeof(int));
  int*   elist = (int*)carve((size_t)(EE + NN) * sizeof(int));
  float* h1pre = bufA;
  float* agg1b = bufB;                       // becomes h1 after LN+ReLU
  float* h2pre = bufA;                       // reuse: h1pre dead after agg1
  float* agg2b = bufA + (size_t)NN * 128;    // becomes h2 after LN+ReLU

  const int TPB = 256;

  // CSR build (independent of GEMM1; same stream keeps ordering)
  gat_init<<<(NN + TPB - 1) / TPB, TPB, 0, stream>>>(deg, curs);
  gat_count<<<(EE + TPB - 1) / TPB, TPB, 0, stream>>>(dstIdx, deg);
  gat_scan<<<1, 256, 0, stream>>>(deg, start, NN);
  gat_fill<<<(EE + TPB - 1) / TPB, TPB, 0, stream>>>(dstIdx, start, curs, elist);
  gat_selfloop<<<(NN + TPB - 1) / TPB, TPB, 0, stream>>>(start, deg, elist);

  // Layer 1: h1pre = x @ W1   (M=50000, K=128, Nc=512)
  gat_gemm_wmma_f32<<<dim3(512 / 64, NN / 16), 128, 0, stream>>>(x, W1, h1pre, 128, 512);
  gat_attn_coef<<<(NN * 4 + TPB - 1) / TPB, TPB, 0, stream>>>(h1pre, as1w, ad1w, as1, ad1, 4, NN * 4);
  gat_agg1<<<NN, 256, 0, stream>>>(h1pre, as1, ad1, srcIdx, start, deg, elist, b1, agg1b);
  gat_ln_relu<<<NN, 256, 0, stream>>>(agg1b, g1, be1, 512);

  // Layer 2: h2pre = h1 @ W2  (M=50000, K=512, Nc=128)
  gat_gemm_wmma_f32<<<dim3(128 / 64, NN / 16), 128, 0, stream>>>(agg1b, W2, h2pre, 512, 128);
  gat_attn_coef<<<(NN + TPB - 1) / TPB, TPB, 0, stream>>>(h2pre, as2w, ad2w, as2, ad2, 1, NN);
  gat_agg2<<<NN, 128, 0, stream>>>(h2pre, as2, ad2, srcIdx, start, deg, elist, b2, agg2b);
  gat_ln_relu<<<NN, 128, 0, stream>>>(agg2b, g2, be2, 128);

  // Pool + MLP head
  gat_pool_head<<<GG, 128, 0, stream>>>(agg2b, batch, rW1, rb1, rg, rbe, rW2, rb2, out);

  (void)in_sizes; (void)n_in; (void)out_size; (void)ws_size;
}